// MultiHeadAttention_80075370267301
// MI455X (gfx1250) — compile-verified
//
#include <hip/hip_runtime.h>
#include <hip/hip_bf16.h>
#include <math.h>

typedef __attribute__((ext_vector_type(2))) float v2f;
typedef __attribute__((ext_vector_type(8))) float v8f;

#define EMBED 1024
#define NHEAD 16
#define HDIM  64
#define BB    4
#define SS    4096
#define NSEG  16

#define WMMA_F32(a, b, c) \
    __builtin_amdgcn_wmma_f32_16x16x4_f32(false, (a), false, (b), (short)0, (c), false, false)

// ---------------------------------------------------------------------------
// GEMM: C[M,N] = A[M,K] @ W[N,K]^T + bias[N]   (fp32 WMMA, 128x128 block tile)
// Both operands staged k-major (stride 36) so every fragment is one b64 load.
// 8 waves in 4x2: each wave owns a 32x64 strip (2 m-tiles x 4 n-tiles).
// ---------------------------------------------------------------------------
__global__ __launch_bounds__(256) void gemm_bias_wmma(
    const float* __restrict__ A, const float* __restrict__ W,
    const float* __restrict__ bias, float* __restrict__ C,
    int M, int N, int K)
{
    __shared__ float As[128][36];   // A tile [m][k]
    __shared__ float Ws[128][36];   // W tile [n][k]  (direct copy, no transpose)

    const int tid   = threadIdx.x;
    const int lane  = tid & 31;
    const int wave  = tid >> 5;
    const int lmod  = lane & 15;
    const int lhalf = lane >> 4;
    const int wmw   = (wave & 3) * 32;   // wave m offset (0,32,64,96)
    const int wnw   = (wave >> 2) * 64;  // wave n offset (0,64)

    const int m0 = blockIdx.x * 128;
    const int n0 = blockIdx.y * 128;

    const int srow = tid >> 1;           // 0..127
    const int scol = (tid & 1) * 16;     // 0 or 16

    v8f acc[2][4];
    #pragma unroll
    for (int i = 0; i < 2; ++i)
        #pragma unroll
        for (int j = 0; j < 4; ++j) acc[i][j] = (v8f){};

    for (int kk = 0; kk < K; kk += 32) {
        const float4* ag = (const float4*)(A + (size_t)(m0 + srow) * K + kk + scol);
        const float4* wg = (const float4*)(W + (size_t)(n0 + srow) * K + kk + scol);
        float4 av0 = ag[0], av1 = ag[1], av2 = ag[2], av3 = ag[3];
        float4 wv0 = wg[0], wv1 = wg[1], wv2 = wg[2], wv3 = wg[3];
        *(float4*)&As[srow][scol +  0] = av0;
        *(float4*)&As[srow][scol +  4] = av1;
        *(float4*)&As[srow][scol +  8] = av2;
        *(float4*)&As[srow][scol + 12] = av3;
        *(float4*)&Ws[srow][scol +  0] = wv0;
        *(float4*)&Ws[srow][scol +  4] = wv1;
        *(float4*)&Ws[srow][scol +  8] = wv2;
        *(float4*)&Ws[srow][scol + 12] = wv3;
        __syncthreads();

        #pragma unroll
        for (int k4 = 0; k4 < 32; k4 += 4) {
            const int ka = k4 + 2 * lhalf;       // element j=0's k
            v2f a0 = *(const v2f*)&As[wmw + lmod][ka];
            v2f a1 = *(const v2f*)&As[wmw + 16 + lmod][ka];
            v2f b0 = *(const v2f*)&Ws[wnw + lmod][ka];
            v2f b1 = *(const v2f*)&Ws[wnw + 16 + lmod][ka];
            v2f b2 = *(const v2f*)&Ws[wnw + 32 + lmod][ka];
            v2f b3 = *(const v2f*)&Ws[wnw + 48 + lmod][ka];
            acc[0][0] = WMMA_F32(a0, b0, acc[0][0]);
            acc[0][1] = WMMA_F32(a0, b1, acc[0][1]);
            acc[0][2] = WMMA_F32(a0, b2, acc[0][2]);
            acc[0][3] = WMMA_F32(a0, b3, acc[0][3]);
            acc[1][0] = WMMA_F32(a1, b0, acc[1][0]);
            acc[1][1] = WMMA_F32(a1, b1, acc[1][1]);
            acc[1][2] = WMMA_F32(a1, b2, acc[1][2]);
            acc[1][3] = WMMA_F32(a1, b3, acc[1][3]);
        }
        __syncthreads();
    }

    #pragma unroll
    for (int tn = 0; tn < 4; ++tn) {
        const int cn = n0 + wnw + tn * 16 + lmod;
        const float bb = bias[cn];
        #pragma unroll
        for (int tm = 0; tm < 2; ++tm) {
            #pragma unroll
            for (int i = 0; i < 8; ++i) {
                const int m = m0 + wmw + tm * 16 + i + 8 * lhalf;
                C[(size_t)m * N + cn] = acc[tm][tn][i] + bb;
            }
        }
    }
}

// ---------------------------------------------------------------------------
// Partial Gram: gram_part[seg][bh][d][e] = sum_{s in seg} qh[s][d]*kh[s][e]
// grid (NSEG, B*H); each block contracts SS/NSEG = 256 rows of S.
// q/k staged transposed to [d][s] (stride 36) -> b64 fragment loads.
// ---------------------------------------------------------------------------
__global__ __launch_bounds__(256) void qk_partial_wmma(
    const float* __restrict__ q, const float* __restrict__ kmat,
    float* __restrict__ gram_part)
{
    __shared__ float qsT[64][36];   // [d][s_local]
    __shared__ float ksT[64][36];   // [e][s_local]

    const int seg = blockIdx.x;
    const int bh  = blockIdx.y;
    const int b   = bh >> 4;
    const int h   = bh & 15;

    const int tid   = threadIdx.x;
    const int lane  = tid & 31;
    const int wave  = tid >> 5;
    const int lmod  = lane & 15;
    const int lhalf = lane >> 4;
    const int wm  = wave >> 1;            // d tile 0..3
    const int en0 = (wave & 1) * 2;       // e tiles en0, en0+1

    const int srow = tid >> 3;            // 0..31 (s within chunk)
    const int dcol = (tid & 7) * 8;       // 0..56 (d)

    v8f acc0 = {}; v8f acc1 = {};

    const int sbeg = seg * (SS / NSEG);
    for (int s0 = sbeg; s0 < sbeg + SS / NSEG; s0 += 32) {
        const float4* qg = (const float4*)(q    + (size_t)(b * SS + s0 + srow) * EMBED + h * HDIM + dcol);
        const float4* kg = (const float4*)(kmat + (size_t)(b * SS + s0 + srow) * EMBED + h * HDIM + dcol);
        float4 q0 = qg[0], q1 = qg[1];
        float4 k0 = kg[0], k1 = kg[1];
        qsT[dcol + 0][srow] = q0.x; qsT[dcol + 1][srow] = q0.y;
        qsT[dcol + 2][srow] = q0.z; qsT[dcol + 3][srow] = q0.w;
        qsT[dcol + 4][srow] = q1.x; qsT[dcol + 5][srow] = q1.y;
        qsT[dcol + 6][srow] = q1.z; qsT[dcol + 7][srow] = q1.w;
        ksT[dcol + 0][srow] = k0.x; ksT[dcol + 1][srow] = k0.y;
        ksT[dcol + 2][srow] = k0.z; ksT[dcol + 3][srow] = k0.w;
        ksT[dcol + 4][srow] = k1.x; ksT[dcol + 5][srow] = k1.y;
        ksT[dcol + 6][srow] = k1.z; ksT[dcol + 7][srow] = k1.w;
        __syncthreads();

        #pragma unroll
        for (int k4 = 0; k4 < 32; k4 += 4) {
            const int ka = k4 + 2 * lhalf;
            v2f a  = *(const v2f*)&qsT[wm * 16 + lmod][ka];
            v2f b0 = *(const v2f*)&ksT[en0 * 16 + lmod][ka];
            v2f b1 = *(const v2f*)&ksT[(en0 + 1) * 16 + lmod][ka];
            acc0 = WMMA_F32(a, b0, acc0);
            acc1 = WMMA_F32(a, b1, acc1);
        }
        __syncthreads();
    }

    float* gp = gram_part + ((size_t)seg * (BB * NHEAD) + bh) * (HDIM * HDIM);
    #pragma unroll
    for (int i = 0; i < 8; ++i) {
        const int d = wm * 16 + i + 8 * lhalf;
        gp[d * 64 + en0 * 16 + lmod]       = acc0[i];
        gp[d * 64 + (en0 + 1) * 16 + lmod] = acc1[i];
    }
}

// ---------------------------------------------------------------------------
// Reduce NSEG partials, softmax rows (scale 1/32), write TRANSPOSED scores:
// scoresT[bh][e][d]  so the AV kernel can stage its B operand k(d)-major.
// grid 64 (bh), 64 threads (one per row d). Deterministic reduction order.
// ---------------------------------------------------------------------------
__global__ __launch_bounds__(64) void gram_softmax_t(
    const float* __restrict__ gram_part, float* __restrict__ scoresT)
{
    const int bh = blockIdx.x;
    const int d  = threadIdx.x;

    float row[64];
    #pragma unroll
    for (int e = 0; e < 64; ++e) row[e] = 0.0f;
    for (int seg = 0; seg < NSEG; ++seg) {
        const float* gp = gram_part + ((size_t)seg * (BB * NHEAD) + bh) * (HDIM * HDIM) + d * 64;
        #pragma unroll
        for (int e = 0; e < 64; ++e) row[e] += gp[e];
    }

    const float scl = 1.0f / 32.0f;          // 1/sqrt(EMBED)
    float mx = -INFINITY;
    #pragma unroll
    for (int e = 0; e < 64; ++e) mx = fmaxf(mx, row[e] * scl);
    float sum = 0.0f;
    #pragma unroll
    for (int e = 0; e < 64; ++e) { row[e] = __expf(row[e] * scl - mx); sum += row[e]; }
    const float inv = 1.0f / sum;
    float* dst = scoresT + (size_t)bh * (HDIM * HDIM);
    #pragma unroll
    for (int e = 0; e < 64; ++e) dst[e * 64 + d] = row[e] * inv;
}

// ---------------------------------------------------------------------------
// merged[b, s, h*64+e] = sum_d vp[b, s, h*64+d] * scores[d][e]
// A = vp tile [s][d] (k-major already), B = scoresT [e][d] (k-major already).
// grid (S/64, B*H), K=64 fully staged, stride 68.
// ---------------------------------------------------------------------------
__global__ __launch_bounds__(256) void av_wmma(
    const float* __restrict__ vp, const float* __restrict__ scoresT,
    float* __restrict__ out)
{
    __shared__ float As[64][68];   // vp tile  [s][d]
    __shared__ float Bs[64][68];   // scoresT  [e][d]

    const int st = blockIdx.x;
    const int bh = blockIdx.y;
    const int b  = bh >> 4;
    const int h  = bh & 15;
    const int s0 = st * 64;

    const int tid   = threadIdx.x;
    const int lane  = tid & 31;
    const int wave  = tid >> 5;
    const int lmod  = lane & 15;
    const int lhalf = lane >> 4;
    const int wm = wave >> 1;
    const int wn = wave & 1;

    const int row = tid >> 2;            // 0..63
    const int col = (tid & 3) * 16;      // 0,16,32,48

    {
        const float4* vg = (const float4*)(vp + (size_t)(b * SS + s0 + row) * EMBED + h * HDIM + col);
        const float4* sg = (const float4*)(scoresT + (size_t)bh * (HDIM * HDIM) + row * 64 + col);
        float4 v0 = vg[0], v1 = vg[1], v2 = vg[2], v3 = vg[3];
        float4 t0 = sg[0], t1 = sg[1], t2 = sg[2], t3 = sg[3];
        *(float4*)&As[row][col +  0] = v0;
        *(float4*)&As[row][col +  4] = v1;
        *(float4*)&As[row][col +  8] = v2;
        *(float4*)&As[row][col + 12] = v3;
        *(float4*)&Bs[row][col +  0] = t0;
        *(float4*)&Bs[row][col +  4] = t1;
        *(float4*)&Bs[row][col +  8] = t2;
        *(float4*)&Bs[row][col + 12] = t3;
    }
    __syncthreads();

    v8f acc0 = {}; v8f acc1 = {};
    #pragma unroll
    for (int k4 = 0; k4 < 64; k4 += 4) {
        const int ka = k4 + 2 * lhalf;
        v2f a  = *(const v2f*)&As[wm * 16 + lmod][ka];
        v2f b0 = *(const v2f*)&Bs[wn * 32 + lmod][ka];
        v2f b1 = *(const v2f*)&Bs[wn * 32 + 16 + lmod][ka];
        acc0 = WMMA_F32(a, b0, acc0);
        acc1 = WMMA_F32(a, b1, acc1);
    }

    const int cb = h * HDIM + wn * 32 + lmod;
    #pragma unroll
    for (int i = 0; i < 8; ++i) {
        const int s = s0 + wm * 16 + i + 8 * lhalf;
        out[(size_t)(b * SS + s) * EMBED + cb]      = acc0[i];
        out[(size_t)(b * SS + s) * EMBED + cb + 16] = acc1[i];
    }
}

// ---------------------------------------------------------------------------
extern "C" void kernel_launch(void* const* d_in, const int* in_sizes, int n_in,
                              void* d_out, int out_size, void* d_ws, size_t ws_size,
                              hipStream_t stream) {
    (void)in_sizes; (void)n_in; (void)out_size; (void)ws_size;
    const float* q  = (const float*)d_in[0];
    const float* k  = (const float*)d_in[1];
    const float* v  = (const float*)d_in[2];
    const float* Wv = (const float*)d_in[3];
    const float* bv = (const float*)d_in[4];
    const float* Wo = (const float*)d_in[5];
    const float* bo = (const float*)d_in[6];
    float* out = (float*)d_out;

    const size_t ME = (size_t)BB * SS * EMBED;            // 16,777,216 floats
    float* vp        = (float*)d_ws;                       // [B*S, E]      64 MB
    float* merged    = vp + ME;                            // [B*S, E]      64 MB
    float* gram_part = merged + ME;                        // [NSEG,BH,64,64] 16 MB
    float* scoresT   = gram_part + (size_t)NSEG * BB * NHEAD * HDIM * HDIM;  // 1 MB

    const int M = BB * SS;                                 // 16384
    dim3 ggrid(M / 128, EMBED / 128);                      // (128, 8)

    // 1) vp = v @ Wv^T + bv
    gemm_bias_wmma<<<ggrid, 256, 0, stream>>>(v, Wv, bv, vp, M, EMBED, EMBED);
    // 2a) partial Gram matrices over S segments
    qk_partial_wmma<<<dim3(NSEG, BB * NHEAD), 256, 0, stream>>>(q, k, gram_part);
    // 2b) reduce + softmax + transpose
    gram_softmax_t<<<BB * NHEAD, 64, 0, stream>>>(gram_part, scoresT);
    // 3) merged = vh @ scores
    av_wmma<<<dim3(SS / 64, BB * NHEAD), 256, 0, stream>>>(vp, scoresT, merged);
    // 4) out = merged @ Wo^T + bo
    gemm_bias_wmma<<<ggrid, 256, 0, stream>>>(merged, Wo, bo, out, M, EMBED, EMBED);
}